// ChannelDropout_14851996910142
// MI455X (gfx1250) — compile-verified
//
#include <hip/hip_runtime.h>
#include <hip/hip_bf16.h>
#include <stdint.h>

// ChannelDropout for MI455X (gfx1250).
// Phase 1: per-channel scale (17,600 scalars) -> d_ws.
// Phase 2: 422 MB stream out[i] = x[i] * scale[i / T]; HBM-bound (~18 us @ 23.3 TB/s).
//          gfx1250 async global->LDS b128 copies (ASYNCcnt), 4-deep pipelined per wave,
//          non-temporal hints on both once-touched streams (422 MB > 192 MB L2).

typedef float v4f __attribute__((ext_vector_type(4)));

#define TPB   256
#define ITERS 16
#define BUFS  4      // async pipeline depth (in-order completion on ASYNCcnt)

// ---------------- Phase 2: HBM stream with async-to-LDS staging ----------------
__global__ __launch_bounds__(TPB) void cd_apply_kernel(
    const float* __restrict__ x,       // [total4 * 4] floats
    const float* __restrict__ scale,   // [nch]
    float* __restrict__ out,
    uint32_t total4,                   // number of float4 elements
    uint64_t magic) {                  // ch = (cidx * magic) >> 40  (== cidx / (T/4))
  __shared__ v4f smem[BUFS * TPB];     // 16 KB: four buffers, one slot per thread

  const uint32_t tid  = threadIdx.x;
  const uint32_t base = blockIdx.x * (TPB * ITERS) + tid;
  const uint64_t xb   = (uint64_t)(uintptr_t)x;

  // LDS byte offsets for this thread's buffer slots (flat addr low 32 bits == LDS offset)
  uint32_t ldsb[BUFS];
#pragma unroll
  for (int b = 0; b < BUFS; ++b)
    ldsb[b] = (uint32_t)(uintptr_t)(&smem[b * TPB + tid]);

  // Prologue: issue BUFS-1 async loads (clamped addresses: tail lanes never fault)
#pragma unroll
  for (int p = 0; p < BUFS - 1; ++p) {
    uint32_t idx = base + (uint32_t)p * TPB;
    uint32_t cl  = idx < total4 ? idx : (total4 - 1u);
    uint64_t ga  = xb + (uint64_t)cl * 16u;
    asm volatile("global_load_async_to_lds_b128 %0, %1, off th:TH_LOAD_NT"
                 :: "v"(ldsb[p]), "v"(ga) : "memory");
  }

#pragma unroll
  for (int i = 0; i < ITERS; ++i) {
    if (i + (BUFS - 1) < ITERS) {
      // Steady state: top up the pipe, then wait for the oldest (== this iter's buffer).
      uint32_t nidx = base + (uint32_t)(i + BUFS - 1) * TPB;
      uint32_t ncl  = nidx < total4 ? nidx : (total4 - 1u);
      uint64_t nga  = xb + (uint64_t)ncl * 16u;
      asm volatile("global_load_async_to_lds_b128 %0, %1, off th:TH_LOAD_NT"
                   :: "v"(ldsb[(i + BUFS - 1) & (BUFS - 1)]), "v"(nga) : "memory");
      asm volatile("s_wait_asynccnt 0x3" ::: "memory");
    } else if (ITERS - 1 - i == 2) {            // drain: 3 outstanding -> need 2 left
      asm volatile("s_wait_asynccnt 0x2" ::: "memory");
    } else if (ITERS - 1 - i == 1) {
      asm volatile("s_wait_asynccnt 0x1" ::: "memory");
    } else {
      asm volatile("s_wait_asynccnt 0x0" ::: "memory");
    }

    uint32_t cidx = base + (uint32_t)i * TPB;
    if (cidx < total4) {
      v4f v = smem[((uint32_t)i & (BUFS - 1)) * TPB + tid];         // ds_read_b128
      uint32_t ch = (uint32_t)(((uint64_t)cidx * magic) >> 40);     // cidx / (T/4)
      float s = scale[ch];                                          // cached (70 KB table)
      v *= s;
      __builtin_nontemporal_store(v, ((v4f*)out) + cidx);           // NT store (streamed once)
    }
  }
}

// ---------------- Phase 1: per-channel gate ----------------
__global__ __launch_bounds__(256) void cd_scale_kernel(
    const float* __restrict__ pos,     // [nch, 2]
    const float* __restrict__ cen,     // [ncen, 2]; row 0 = ban center
    float* __restrict__ scale,         // [nch] out
    int nch, int ncen, float inv_mc) { // inv_mc = 1/(ncen-1)
  int j = blockIdx.x * blockDim.x + threadIdx.x;
  if (j >= nch) return;

  float px = pos[2 * j + 0];
  float py = pos[2 * j + 1];
  bool valid = !((px == -2.0f) && (py == -2.0f));   // INVALID sentinel

  const float R2 = 0.1f * 0.1f;                     // DROPOUT^2 (dist>R <=> d2>R2)
  float dx0 = px - cen[0];
  float dy0 = py - cen[1];
  bool kept0 = (dx0 * dx0 + dy0 * dy0) > R2;

  float cnt = 0.0f;
  for (int n = 1; n < ncen; ++n) {
    float dx = px - cen[2 * n + 0];
    float dy = py - cen[2 * n + 1];
    cnt += ((dx * dx + dy * dy) > R2) ? 1.0f : 0.0f;
  }
  // invalid channels: x already zeroed -> scale 0 regardless of prob
  float prob = cnt * inv_mc;
  float s = (valid && kept0) ? (1.0f / (prob + 1e-8f)) : 0.0f;
  scale[j] = s;
}

extern "C" void kernel_launch(void* const* d_in, const int* in_sizes, int n_in,
                              void* d_out, int out_size, void* d_ws, size_t ws_size,
                              hipStream_t stream) {
  const float* x   = (const float*)d_in[0];   // [B, C, T]
  const float* pos = (const float*)d_in[1];   // [B, C, 2]
  const float* cen = (const float*)d_in[2];   // [ncen, 2]
  float* out   = (float*)d_out;
  float* scale = (float*)d_ws;                // [B*C] floats of scratch

  const int nch  = in_sizes[1] / 2;           // B*C = 17,600
  const int T    = in_sizes[0] / nch;         // 3000 (divisible by 4)
  const int ncen = in_sizes[2] / 2;           // 101
  const uint32_t t4     = (uint32_t)(T / 4);            // 750 float4 per channel
  const uint32_t total4 = (uint32_t)(in_sizes[0] / 4);  // 13.2M float4
  const uint64_t magic  = ((1ULL << 40) + t4 - 1) / t4; // exact for idx < ~1.4e9
  const float inv_mc = 1.0f / (float)(ncen - 1);

  dim3 b1(256), g1((nch + 255) / 256);
  cd_scale_kernel<<<g1, b1, 0, stream>>>(pos, cen, scale, nch, ncen, inv_mc);

  const uint32_t per_block = TPB * ITERS;     // 4096 float4 (64 KB) per block
  const uint32_t nblocks = (total4 + per_block - 1) / per_block;
  cd_apply_kernel<<<nblocks, TPB, 0, stream>>>(x, scale, out, total4, magic);
}